// FMAdam_56788057588236
// MI455X (gfx1250) — compile-verified
//
#include <hip/hip_runtime.h>
#include <hip/hip_bf16.h>

typedef __attribute__((ext_vector_type(2))) float f32x2;
typedef __attribute__((ext_vector_type(4))) float f32x4;
typedef __attribute__((ext_vector_type(8))) float f32x8;

#define NUM_FIELDS 39
#define FIELD_VOCAB 100000LL
#define EMBED_DIM 16
#define BATCH 16384

// One wave (32 lanes) handles a 16-sample tile: 2 lanes per sample, 8 embedding
// dims per lane (two b128 gathers per field). 8 waves / block -> 128 samples.
__global__ __launch_bounds__(256) void fm_fwd_kernel(
    const long long* __restrict__ Xi,      // [B, F] int64
    const float*     __restrict__ Xv,      // [B, F]
    const float*     __restrict__ first,   // [F*V, 1]
    const float*     __restrict__ second,  // [F*V, 16]
    const float*     __restrict__ bias,    // [1]
    float*           __restrict__ out)     // [B]
{
    __shared__ float S[8][16][16];  // per-wave 16x16 fp32 tile: S[sample][dim]
    __shared__ float D[8][16][16];  // per-wave S*S^T result
    __shared__ float T[8][16];      // per-sample (fo - 0.5*ssq)

    const int tid  = threadIdx.x;
    const int w    = tid >> 5;          // wave in block
    const int lane = tid & 31;
    const int s    = lane >> 1;         // sample within wave tile (0..15)
    const int h    = lane & 1;          // dim half: dims [8h, 8h+8)
    const long long b = (long long)blockIdx.x * 128 + w * 16 + s;
    const long long rowXi = b * NUM_FIELDS;

    f32x4 a0 = {0.f, 0.f, 0.f, 0.f};
    f32x4 a1 = {0.f, 0.f, 0.f, 0.f};
    float ssq = 0.f, fo = 0.f;

    #pragma unroll 3
    for (int f = 0; f < NUM_FIELDS; ++f) {
        // prefetch the row 4 fields ahead to break the idx->row latency chain
        if (f + 4 < NUM_FIELDS) {
            long long pidx = Xi[rowXi + f + 4] + (long long)(f + 4) * FIELD_VOCAB;
            __builtin_prefetch(second + pidx * EMBED_DIM, 0, 0);
        }
        long long idx = Xi[rowXi + f] + (long long)f * FIELD_VOCAB;
        float xv = Xv[rowXi + f];

        const f32x4* row = (const f32x4*)(second + idx * EMBED_DIM) + 2 * h;
        f32x4 q0 = row[0];
        f32x4 q1 = row[1];

        a0 += xv * q0;
        a1 += xv * q1;
        float d2 = q0.x*q0.x + q0.y*q0.y + q0.z*q0.z + q0.w*q0.w
                 + q1.x*q1.x + q1.y*q1.y + q1.z*q1.z + q1.w*q1.w;
        ssq += xv * xv * d2;

        // first-order term: split fields between the two lanes of the pair
        if ((f & 1) == h) fo += first[idx] * xv;
    }

    // reduce ssq/fo across the lane pair of each sample (wave32 shuffle)
    ssq += __shfl_xor(ssq, 1, 32);
    fo  += __shfl_xor(fo,  1, 32);

    // stage S tile to LDS: S[s][dim]
    f32x4* srow = (f32x4*)&S[w][s][8 * h];
    srow[0] = a0;
    srow[1] = a1;
    if (h == 0) T[w][s] = fo - 0.5f * ssq;
    __syncthreads();

    // D = S * S^T via V_WMMA_F32_16X16X4_F32, K = 16 in 4 steps.
    // A layout (32-bit 16x4, ISA 7.12.2): lanes 0-15 -> M=lane, K={4k,4k+1};
    // lanes 16-31 -> M=lane-16, K={4k+2,4k+3}. B = S^T under the symmetric
    // striping means the B operand registers equal the A operand registers.
    {
        const int M  = lane & 15;
        const int kk = (lane >> 4) << 1;  // 0 or 2
        f32x8 c = {0.f, 0.f, 0.f, 0.f, 0.f, 0.f, 0.f, 0.f};
        #pragma unroll
        for (int k = 0; k < 4; ++k) {
            f32x2 ab = *(const f32x2*)&S[w][M][4 * k + kk];  // 8B-aligned
            c = __builtin_amdgcn_wmma_f32_16x16x4_f32(
                    /*neg_a=*/false, ab, /*neg_b=*/false, ab,
                    /*c_mod=*/(short)0, c, /*reuse_a=*/false, /*reuse_b=*/false);
        }
        // C/D layout: VGPR r @ lanes 0-15 -> (M=r, N=lane); lanes 16-31 -> (M=r+8, N=lane-16)
        const int mhi = (lane >> 4) << 3;
        #pragma unroll
        for (int r = 0; r < 8; ++r)
            D[w][r + mhi][M] = c[r];
    }
    __syncthreads();

    // out[b] = fo - 0.5*ssq + 0.5*||s||^2 + bias ; diagonal read is stride-17
    // words -> conflict-free across 64 banks.
    if (lane < 16) {
        long long bb = (long long)blockIdx.x * 128 + w * 16 + lane;
        out[bb] = T[w][lane] + 0.5f * D[w][lane][lane] + bias[0];
    }
}

extern "C" void kernel_launch(void* const* d_in, const int* in_sizes, int n_in,
                              void* d_out, int out_size, void* d_ws, size_t ws_size,
                              hipStream_t stream) {
    const long long* Xi     = (const long long*)d_in[0];
    const float*     Xv     = (const float*)d_in[1];
    const float*     first  = (const float*)d_in[2];
    const float*     second = (const float*)d_in[3];
    const float*     bias   = (const float*)d_in[4];
    float*           out    = (float*)d_out;

    dim3 grid(BATCH / 128);   // 128 samples per block
    dim3 block(256);          // 8 wave32 waves
    fm_fwd_kernel<<<grid, block, 0, stream>>>(Xi, Xv, first, second, bias, out);
}